// FeatureCalculator_89343909691675
// MI455X (gfx1250) — compile-verified
//
#include <hip/hip_runtime.h>
#include <math.h>

typedef __attribute__((ext_vector_type(2))) float v2f;
typedef __attribute__((ext_vector_type(8))) float v8f;

#define SR      48000
#define N_FFT   960
#define N_HOP   480
#define N_ERB   32
#define N_DF    96
#define F_BINS  481
#define F_PAD   496          /* 31 tiles of 16 bins */
#define N_T     4000         /* frames kept (last dropped) */
#define N_CH    8
#define SPAN    8160         /* 15*480 + 960 samples per 16-frame tile */
#define ALPHA   0.99f
#define TWO_PI_OVER_N (6.2831853071795864769f / 960.0f)
#define HALF_PI 1.5707963267948966f

/* -------- workspace layout (float units) -------- */
/* Packed twiddle table: TW[k/2][f] = {cos(k,f), cos(k+1,f), sin(k,f), sin(k+1,f)}
 * -> one b128 load supplies both B fragments (cos+sin) for a K-pair.     */
#define WS_WIN   0
#define WS_TW    (WS_WIN + N_FFT)                 /* 960; 480*496*4 floats */
#define WS_BAND  (WS_TW + (N_FFT / 2) * F_PAD * 4) /* ints, 496            */
#define WS_WINV  (WS_BAND + F_PAD)                /* 32 floats             */
#define WS_ACC   (WS_WINV + N_ERB)                /* 8*4000*32 floats      */

/* -------- output layout (float units) -------- */
#define OUT_SPEC 0
#define OUT_ERB  (N_CH * N_T * F_BINS * 2)        /* 30,784,000 */
#define OUT_DF   (OUT_ERB + N_CH * N_T * N_ERB)   /* 31,808,000 */

/* ============================================================
 * Kernel A: window, packed DFT twiddles, ERB band map.
 * Integer phase reduction (k*f mod 960) keeps fp32 sincos exact-ish.
 * ============================================================ */
__global__ __launch_bounds__(256) void k_tables(float* __restrict__ ws) {
    const int tid = threadIdx.x;

    for (int i = tid; i < N_FFT; i += 256) {
        float s = __sinf(HALF_PI * ((float)i + 0.5f) / 480.0f);
        ws[WS_WIN + i] = __sinf(HALF_PI * s * s);
    }

    float4* tw = (float4*)(ws + WS_TW);
    for (int idx = tid; idx < (N_FFT / 2) * F_PAD; idx += 256) {
        int kb = idx / F_PAD;            /* K-pair index: rows 2kb, 2kb+1 */
        int f  = idx - kb * F_PAD;
        int k0 = kb * 2;
        int p0 = (int)(((long long)k0 * (long long)f) % N_FFT);
        int p1 = (int)(((long long)(k0 + 1) * (long long)f) % N_FFT);
        float a0 = TWO_PI_OVER_N * (float)p0;
        float a1 = TWO_PI_OVER_N * (float)p1;
        float4 v;
        v.x = __cosf(a0); v.y = __cosf(a1);
        v.z = __sinf(a0); v.w = __sinf(a1);
        tw[idx] = v;
    }

    if (tid == 0) {
        int* band = (int*)(ws + WS_BAND);
        const double c = 24.7 * 9.265;
        double ehigh = 9.265 * log1p((SR * 0.5) / c);
        double stepe = ehigh / (double)N_ERB;
        double fw = (double)SR / (double)N_FFT;
        int widths[N_ERB];
        int prev = 0, over = 0;
        for (int i = 1; i <= N_ERB; ++i) {
            double fr = c * (exp((double)i * stepe / 9.265) - 1.0);
            int fb = (int)floor(fr / fw + 0.5);
            int nb = fb - prev - over;
            if (nb < 2) { over = 2 - nb; nb = 2; } else { over = 0; }
            widths[i - 1] = nb;
            prev = fb;
        }
        widths[N_ERB - 1] += 1;
        int total = 0;
        for (int i = 0; i < N_ERB; ++i) total += widths[i];
        if (total > F_BINS) widths[N_ERB - 1] -= (total - F_BINS);

        int b = 0, cnt = 0;
        for (int i = 0; i < F_PAD; ++i) {
            if (i < F_BINS) {
                band[i] = b;
                if (++cnt >= widths[b] && b < N_ERB - 1) { b++; cnt = 0; }
            } else {
                band[i] = N_ERB - 1;
            }
        }
        for (int i = 0; i < N_ERB; ++i) ws[WS_WINV + i] = 1.0f / (float)widths[i];
    }
}

/* ============================================================
 * Kernel B: fused STFT (DFT as WMMA f32 16x16x4) + power + ERB sums.
 * Block = 1 channel x 16 frames; 8 waves stride the 31 bin-tiles.
 * Samples staged via global_load_async_to_lds_b32 (ASYNCcnt path).
 * Inner loop: 1x global_load_b128 (packed twiddles) + LDS b64 pair
 * + 2x v_wmma per K-step of 4.
 * ============================================================ */
__global__ __launch_bounds__(256) void k_stft(const float* __restrict__ x,
                                              const float* __restrict__ ws,
                                              float* __restrict__ acc,
                                              float* __restrict__ out) {
    __shared__ float lds_x[SPAN];
    __shared__ float lds_win[N_FFT];
    __shared__ float lds_erb[16][N_ERB];

    const int tid = threadIdx.x;
    const int c   = blockIdx.y;
    const int t0  = blockIdx.x * 16;
    const float* xc = x + (size_t)c * (size_t)(N_T * N_HOP);

    for (int i = tid; i < 16 * N_ERB; i += 256) ((float*)lds_erb)[i] = 0.0f;
    for (int i = tid; i < N_FFT; i += 256)      lds_win[i] = ws[WS_WIN + i];

    /* async stage of the 8160-sample span; reflect-pad only hits tile 0 */
    for (int i = tid; i < SPAN; i += 256) {
        long g = (long)t0 * N_HOP - N_HOP + (long)i;
        if (g < 0) g = -g;                       /* reflect */
        const float* gp = xc + g;
        unsigned lo = (unsigned)(size_t)(&lds_x[i]);
        asm volatile("global_load_async_to_lds_b32 %0, %1, off"
                     :: "v"(lo), "v"(gp) : "memory");
    }
    asm volatile("s_wait_asynccnt 0" ::: "memory");
    __syncthreads();

    const int wave = tid >> 5;
    const int lane = tid & 31;
    const int mA   = lane & 15;              /* A row (frame within tile)  */
    const int kh   = (lane >> 4) << 1;       /* A/B K sub-offset: 0 or 2   */
    const int nlan = lane & 15;              /* B/D column                 */
    const int mD0  = (lane >> 4) << 3;       /* D row base: 0 or 8         */

    const float4* tw  = (const float4*)(ws + WS_TW);
    const int*   band = (const int*)(ws + WS_BAND);
    const float* xrow = &lds_x[mA * N_HOP];

    for (int nt = wave; nt < 31; nt += 8) {
        const int nb = nt * 16;
        v8f cre = {0.f,0.f,0.f,0.f,0.f,0.f,0.f,0.f};
        v8f cim = {0.f,0.f,0.f,0.f,0.f,0.f,0.f,0.f};

        const float4* twc = tw + nb + nlan + ((size_t)(kh >> 1)) * F_PAD;

        for (int k = 0; k < N_FFT; k += 4) {
            const int k0 = k + kh;
            v2f a;
            a.x = lds_win[k0]     * xrow[k0];
            a.y = lds_win[k0 + 1] * xrow[k0 + 1];

            /* packed B: {cos(k0), cos(k0+1), sin(k0), sin(k0+1)} at col n */
            const float4 w = twc[(size_t)(k >> 1) * F_PAD];
            v2f bc; bc.x = w.x; bc.y = w.y;
            v2f bs; bs.x = w.z; bs.y = w.w;

            cre = __builtin_amdgcn_wmma_f32_16x16x4_f32(
                    false, a, false, bc, (short)0, cre, false, false);
            cim = __builtin_amdgcn_wmma_f32_16x16x4_f32(
                    false, a, false, bs, (short)0, cim, false, false);
        }

        const int n = nb + nlan;
        if (n < F_BINS) {
            const int bd = band[n];
            float2* sp2 = (float2*)(out + OUT_SPEC);
            #pragma unroll
            for (int r = 0; r < 8; ++r) {
                const int mm = mD0 + r;
                const float re =  cre[r];
                const float im = -cim[r];          /* e^{-i theta} */
                const size_t oi = (((size_t)c * N_T + (t0 + mm)) * F_BINS) + n;
                float2 v; v.x = re; v.y = im;
                sp2[oi] = v;                        /* global_store_b64 */
                atomicAdd(&lds_erb[mm][bd], re * re + im * im);
            }
        }
    }
    __syncthreads();

    for (int i = tid; i < 16 * N_ERB; i += 256) {
        int mm = i >> 5, bd = i & 31;
        acc[(((size_t)c * N_T) + t0 + mm) * N_ERB + bd] = ((float*)lds_erb)[i];
    }
}

/* ============================================================
 * Kernel C: ERB dB + sequential EMA normalization over T.
 * ============================================================ */
__global__ __launch_bounds__(256) void k_erb_scan(const float* __restrict__ acc,
                                                  const float* __restrict__ winv,
                                                  float* __restrict__ out) {
    const int tid = threadIdx.x;
    const int c = tid >> 5, b = tid & 31;
    const float wi = winv[b];
    float s = -60.0f - 30.0f * (float)b / 31.0f;      /* linspace(-60,-90,32) */
    const size_t base = ((size_t)c * N_T) * N_ERB + b;
    for (int t = 0; t < N_T; ++t) {
        const size_t idx = base + (size_t)t * N_ERB;
        if (t + 8 < N_T) __builtin_prefetch(&acc[idx + 8 * N_ERB], 0, 0);
        float xdb = 10.0f * __log10f(acc[idx] * wi + 1e-10f);
        s = (1.0f - ALPHA) * xdb + ALPHA * s;
        out[OUT_ERB + idx] = (xdb - s) * (1.0f / 40.0f);
    }
}

/* ============================================================
 * Kernel D: DF-spec unit-norm: s = |x|(1-a) + a*s ; y = x/sqrt(s)
 * ============================================================ */
__global__ __launch_bounds__(256) void k_df_scan(float* __restrict__ out) {
    const int gid = blockIdx.x * blockDim.x + threadIdx.x;
    if (gid >= N_CH * N_DF) return;
    const int c = gid / N_DF, f = gid - c * N_DF;
    float s = 0.001f - 0.0009f * (float)f / 95.0f;    /* linspace(1e-3,1e-4,96) */
    for (int t = 0; t < N_T; ++t) {
        const size_t si = ((((size_t)c * N_T) + t) * F_BINS + f) * 2;
        if (t + 8 < N_T) __builtin_prefetch(&out[si + (size_t)8 * F_BINS * 2], 0, 0);
        const float re = out[OUT_SPEC + si];
        const float im = out[OUT_SPEC + si + 1];
        s = __fsqrt_rn(re * re + im * im) * (1.0f - ALPHA) + s * ALPHA;
        const float inv = __frsqrt_rn(s);
        const size_t di = ((((size_t)c * N_T) + t) * N_DF + f) * 2;
        out[OUT_DF + di]     = re * inv;
        out[OUT_DF + di + 1] = im * inv;
    }
}

extern "C" void kernel_launch(void* const* d_in, const int* in_sizes, int n_in,
                              void* d_out, int out_size, void* d_ws, size_t ws_size,
                              hipStream_t stream) {
    const float* x  = (const float*)d_in[0];
    float* out = (float*)d_out;
    float* ws  = (float*)d_ws;

    k_tables<<<1, 256, 0, stream>>>(ws);

    dim3 grid(N_T / 16, N_CH);
    k_stft<<<grid, 256, 0, stream>>>(x, ws, ws + WS_ACC, out);

    k_erb_scan<<<1, 256, 0, stream>>>(ws + WS_ACC, ws + WS_WINV, out);
    k_df_scan<<<3, 256, 0, stream>>>(out);
}